// ProcessFeatures_83296595738632
// MI455X (gfx1250) — compile-verified
//
#include <hip/hip_runtime.h>
#include <math.h>

typedef __attribute__((ext_vector_type(2))) float v2f;
typedef __attribute__((ext_vector_type(8))) float v8f;

#define BS 96
#define NCH 4
#define W 64
#define C 16
#define L 4096            // NCH*W*C elements per image
#define TILE 16
#define WAVES 16
#define PPW 4             // shifts (p) per wave; WAVES*PPW == 64
#define SSTRIDE 1028      // padded LDS row stride in floats (1024 + 4) -> conflict-free A/B loads

// ---------------------------------------------------------------------------
// CDNA5 async global->LDS copy (ASYNCcnt-tracked, bypasses VGPRs).
//   LDS byte address from a VGPR; global address = SGPR base + VGPR byte offset
// ---------------------------------------------------------------------------
__device__ __forceinline__ void async_copy_b128(unsigned ldsOffBytes,
                                                const float* base,
                                                unsigned byteOff) {
  asm volatile("global_load_async_to_lds_b128 %0, %1, %2"
               :
               : "v"(ldsOffBytes), "v"(byteOff), "s"(base)
               : "memory");
}

__device__ __forceinline__ void wait_asynccnt0() {
  asm volatile("s_wait_asynccnt 0" ::: "memory");
}

// ---------------------------------------------------------------------------
// norms: invnorm[i] = 1 / sqrt(sum(sat[i]^2) + 1e-8)
// ---------------------------------------------------------------------------
__global__ __launch_bounds__(256) void norm_kernel(const float* __restrict__ sat,
                                                   float* __restrict__ invnorm) {
  __shared__ float red[256];
  const int i = blockIdx.x;
  const float* p = sat + (size_t)i * L;
  float s = 0.0f;
  for (int t = threadIdx.x; t < L; t += 256) {
    float v = p[t];
    s += v * v;
  }
  red[threadIdx.x] = s;
  __syncthreads();
  for (int off = 128; off > 0; off >>= 1) {
    if (threadIdx.x < off) red[threadIdx.x] += red[threadIdx.x + off];
    __syncthreads();
  }
  if (threadIdx.x == 0) invnorm[i] = 1.0f / sqrtf(red[0] + 1e-8f);
}

// ---------------------------------------------------------------------------
// Fused circular-correlation GEMM (WMMA f32 16x16x4) + argmax + distance.
//   corr[i,j,p] = sum_{ch,dh,k} sat[i,ch,(p+dh)%64,k] * grd[j,ch,dh,15-k]
//   C   [i,j,p] = sum_{ch,dh,k} sat[i,ch,(p+dh)%64,k] * grd[j,ch,dh,k]
//   orien = argmax_p corr ; dist[j,i] = 2 - 2*C[i,j,orien]*invnorm[i]
// One block per (16 x 16) (i,j) tile, 16 waves, wave w owns p = 4w..4w+3.
// ---------------------------------------------------------------------------
__global__ __launch_bounds__(512) void corr_wmma_kernel(
    const float* __restrict__ sat, const float* __restrict__ grd,
    const float* __restrict__ invnorm,
    float* __restrict__ dist, int* __restrict__ orien) {
  __shared__ float smem[2 * TILE * SSTRIDE];  // per-channel sat tile | grd tile
  float* satS = smem;
  float* grdS = smem + TILE * SSTRIDE;

  const int i0 = blockIdx.x * TILE;
  const int j0 = blockIdx.y * TILE;
  const int tid = threadIdx.x;
  const int wv = tid >> 5;   // wave id 0..15
  const int ln = tid & 31;   // lane
  const int m  = ln & 15;    // M (A) / N (B) index within tile
  const int hh = ln >> 4;    // lane half selects K pair / M+8
  const int pbase = wv * PPW;

  // LDS byte offsets of the two staging buffers (generic ptr low 32 bits = LDS addr)
  const unsigned satBase = (unsigned)(uintptr_t)(void*)satS;
  const unsigned grdBase = (unsigned)(uintptr_t)(void*)grdS;

  v8f corrAcc[PPW] = {};
  v8f cAcc[PPW] = {};

  for (int ch = 0; ch < NCH; ++ch) {
    __syncthreads();  // previous-iteration LDS reads done
    // Async stage of the (16 x 64 x 16) f32 slabs for this channel: each thread
    // issues 8 b128 transfers per tensor, LDS written directly (no VGPR bounce).
    for (int t = tid; t < TILE * 256; t += 512) {
      const int il = t >> 8;       // tile row 0..15
      const int off4 = t & 255;    // float4 index within 1024-float row
      const unsigned ldsOff = (unsigned)((il * SSTRIDE + off4 * 4) * 4);
      const unsigned gOffS = (unsigned)(((i0 + il) * L + ch * 1024 + off4 * 4) * 4);
      const unsigned gOffG = (unsigned)(((j0 + il) * L + ch * 1024 + off4 * 4) * 4);
      async_copy_b128(satBase + ldsOff, sat, gOffS);
      async_copy_b128(grdBase + ldsOff, grd, gOffG);
    }
    wait_asynccnt0();
    __syncthreads();

    for (int dh = 0; dh < W; ++dh) {
#pragma unroll
      for (int kc = 0; kc < 4; ++kc) {
        const int kk = kc * 4 + 2 * hh;  // this lane's K pair within the 16 channels
        // B (plain) and B (k-reversed) operands, shared across this wave's p values
        v2f b   = *(const v2f*)(grdS + m * SSTRIDE + dh * 16 + kk);
        v2f brl = *(const v2f*)(grdS + m * SSTRIDE + dh * 16 + (14 - kk));
        v2f br;
        br.x = brl.y;  // element 15-kk   -> K = 2*hh
        br.y = brl.x;  // element 14-kk   -> K = 2*hh+1
#pragma unroll
        for (int pp = 0; pp < PPW; ++pp) {
          const int row = (pbase + pp + dh) & 63;  // circular shift
          v2f a = *(const v2f*)(satS + m * SSTRIDE + row * 16 + kk);
          corrAcc[pp] = __builtin_amdgcn_wmma_f32_16x16x4_f32(
              false, a, false, br, (short)0, corrAcc[pp], false, false);
          cAcc[pp] = __builtin_amdgcn_wmma_f32_16x16x4_f32(
              false, a, false, b, (short)0, cAcc[pp], false, false);
        }
      }
    }
  }

  // -------- argmax over p (first-max semantics: ascending scan, strict >) ----
  __syncthreads();
  float* rVal = smem;                          // [WAVES][8][32]
  float* rC   = smem + WAVES * 8 * 32;         // [WAVES][8][32]
  int*   rP   = (int*)(smem + 2 * WAVES * 8 * 32);

#pragma unroll
  for (int v = 0; v < 8; ++v) {
    float bv = corrAcc[0][v];
    float bc = cAcc[0][v];
    int bp = pbase;
#pragma unroll
    for (int pp = 1; pp < PPW; ++pp) {
      float cv = corrAcc[pp][v];
      if (cv > bv) { bv = cv; bc = cAcc[pp][v]; bp = pbase + pp; }
    }
    const int slot = (wv * 8 + v) * 32 + ln;
    rVal[slot] = bv;
    rC[slot] = bc;
    rP[slot] = bp;
  }
  __syncthreads();

  if (wv < 8) {  // thread (wv, ln) finalizes accumulator row v = wv
    float bv = rVal[wv * 32 + ln];
    float bc = rC[wv * 32 + ln];
    int bp = rP[wv * 32 + ln];
#pragma unroll
    for (int w2 = 1; w2 < WAVES; ++w2) {
      const int s = (w2 * 8 + wv) * 32 + ln;
      float cv = rVal[s];
      if (cv > bv) { bv = cv; bc = rC[s]; bp = rP[s]; }
    }
    // C/D layout: VGPR v, lane ln  ->  M = v + 8*(ln>=16), N = ln%16
    const int il = wv + 8 * hh;
    const int jl = m;
    const int gi = i0 + il;
    const int gj = j0 + jl;
    orien[gi * BS + gj] = bp;
    const float inv = invnorm[gi];
    dist[gj * BS + gi] = 2.0f - 2.0f * bc * inv;
  }
}

// ---------------------------------------------------------------------------
extern "C" void kernel_launch(void* const* d_in, const int* in_sizes, int n_in,
                              void* d_out, int out_size, void* d_ws, size_t ws_size,
                              hipStream_t stream) {
  (void)in_sizes; (void)n_in; (void)out_size; (void)ws_size;
  const float* sat = (const float*)d_in[0];
  const float* grd = (const float*)d_in[1];
  float* out = (float*)d_out;
  float* invnorm = (float*)d_ws;  // 96 floats of scratch

  const size_t featElems = (size_t)BS * L;  // 393216
  // Passthrough outputs 0 and 1
  hipMemcpyAsync(out, sat, featElems * sizeof(float), hipMemcpyDeviceToDevice, stream);
  hipMemcpyAsync(out + featElems, grd, featElems * sizeof(float), hipMemcpyDeviceToDevice, stream);

  float* dist = out + 2 * featElems;                   // 96*96 f32, layout [j][i]
  int* orien = (int*)(out + 2 * featElems + BS * BS);  // 96*96 i32, layout [i][j]

  norm_kernel<<<BS, 256, 0, stream>>>(sat, invnorm);

  dim3 grid(BS / TILE, BS / TILE);  // 6 x 6 tiles
  corr_wmma_kernel<<<grid, 512, 0, stream>>>(sat, grd, invnorm, dist, orien);
}